// V2E_layer_47390669144619
// MI455X (gfx1250) — compile-verified
//
#include <hip/hip_runtime.h>

#define N_NODES 1000000
#define N_EDGES 100000
#define D_FEAT  128

typedef float v2f __attribute__((ext_vector_type(2)));
typedef float v8f __attribute__((ext_vector_type(8)));

#define NZERO (N_EDGES * D_FEAT + N_EDGES)
#define NSWZ1 (32 * 8 * 32)   // W_v2e fragments: 32 k-steps x 8 n-tiles x 32 lanes (float2 each)
#define NSWZ2 (64 * 8 * 32)   // W_upd fragments: 64 k-steps x 8 n-tiles x 32 lanes

// ---------------------------------------------------------------------------
// K0a: zero scratch (agg_sum [E*128] + count [E])
// ---------------------------------------------------------------------------
__global__ void zero_ws_kernel(float* __restrict__ p, int n) {
    int i = blockIdx.x * blockDim.x + threadIdx.x;
    int stride = gridDim.x * blockDim.x;
    for (; i < n; i += stride) p[i] = 0.0f;
}

// ---------------------------------------------------------------------------
// K0b: pre-swizzle W_v2e / W_upd into WMMA B-fragment order so the GEMM
// kernels fetch one aligned float2 per lane per (k-step, n-tile).
// Fragment layout per ISA 7.12.2 (32-bit B 4x16): lane L holds N=L&15,
// K = 4*kk + 2*(L>>4) + {0,1}.
// ---------------------------------------------------------------------------
__global__ void swizzle_w_kernel(const float* __restrict__ W1,
                                 const float* __restrict__ W2,
                                 float* __restrict__ b1,
                                 float* __restrict__ b2) {
    int idx = blockIdx.x * blockDim.x + threadIdx.x;
    if (idx >= NSWZ1 + NSWZ2) return;
    if (idx < NSWZ1) {
        const int kk = idx >> 8, nt = (idx >> 5) & 7, lane = idx & 31;
        const int m = lane & 15, hi = lane >> 4;
        const int k0 = 4 * kk + 2 * hi, n = nt * 16 + m;
        v2f v; v.x = W1[k0 * D_FEAT + n]; v.y = W1[(k0 + 1) * D_FEAT + n];
        ((v2f*)b1)[idx] = v;
    } else {
        const int i2 = idx - NSWZ1;
        const int kk = i2 >> 8, nt = (i2 >> 5) & 7, lane = i2 & 31;
        const int m = lane & 15, hi = lane >> 4;
        const int k0 = 4 * kk + 2 * hi, n = nt * 16 + m;
        v2f v; v.x = W2[k0 * D_FEAT + n]; v.y = W2[(k0 + 1) * D_FEAT + n];
        ((v2f*)b2)[i2] = v;
    }
}

// ---------------------------------------------------------------------------
// K1: node_info = relu(hyper_node @ W_v2e + b) fused with scatter-add.
// One wave = 16 rows x 128 cols, WMMA f32 16x16x4, 32 k-steps x 8 n-tiles.
// hyper_node is a 512MB read-once stream -> NT loads (keep L2 for agg_sum).
// ---------------------------------------------------------------------------
__global__ __launch_bounds__(256) void gemm1_scatter_kernel(
    const float* __restrict__ hn,      // [N_NODES,128]
    const v2f*   __restrict__ bswz,    // pre-swizzled W_v2e fragments
    const float* __restrict__ bias,    // [128]
    const long long* __restrict__ seg, // ve_affiliation row 0, [N_NODES]
    float* __restrict__ agg_sum,       // [N_EDGES,128]
    float* __restrict__ cnt)           // [N_EDGES]
{
    const int lane    = threadIdx.x & 31;
    const int wave    = threadIdx.x >> 5;
    const int rowbase = (blockIdx.x * 8 + wave) * 16;
    const int m       = lane & 15;   // A row / B,C,D column within tile
    const int hi      = lane >> 4;   // half-wave select

    v8f acc[8];
#pragma unroll
    for (int t = 0; t < 8; ++t) acc[t] = 0.0f;

    // A fragment source row (clamped for tail; EXEC stays all-ones for WMMA)
    const int rowA = min(rowbase + m, N_NODES - 1);
    const float* __restrict__ arow = hn + (long)rowA * D_FEAT;

    for (int kk = 0; kk < 32; ++kk) {
        const int k0 = 4 * kk + 2 * hi;                 // even -> 8B aligned
        v2f a = __builtin_nontemporal_load((const v2f*)(arow + k0));
        const v2f* __restrict__ bp = bswz + (size_t)(kk * 8) * 32 + lane;
#pragma unroll
        for (int nt = 0; nt < 8; ++nt) {
            v2f b = bp[nt * 32];
            acc[nt] = __builtin_amdgcn_wmma_f32_16x16x4_f32(
                false, a, false, b, (short)0, acc[nt], false, false);
        }
    }

    // hoist bias for this lane's columns
    float bv[8];
#pragma unroll
    for (int nt = 0; nt < 8; ++nt) bv[nt] = bias[nt * 16 + m];

    // bias + relu + scatter-add into L2-resident agg_sum / cnt
#pragma unroll
    for (int j = 0; j < 8; ++j) {
        const int row = rowbase + j + 8 * hi;           // C/D: VGPR j -> M = j + 8*hi
        if (row < N_NODES) {
            const long long e = seg[row];
#pragma unroll
            for (int nt = 0; nt < 8; ++nt) {
                float v = fmaxf(acc[nt][j] + bv[nt], 0.0f);
                unsafeAtomicAdd(&agg_sum[(long)e * D_FEAT + nt * 16 + m], v);
            }
            if (m == 0) unsafeAtomicAdd(&cnt[e], 1.0f);
        }
    }
}

// ---------------------------------------------------------------------------
// K2: upd = relu(concat([agg_sum/clamp(cnt), hyperedge]) @ W_upd + b),
//     then row-wise L2 normalize. Same WMMA tiling over 100k edge rows.
// ---------------------------------------------------------------------------
__global__ __launch_bounds__(256) void gemm2_norm_kernel(
    const float* __restrict__ agg_sum, // [N_EDGES,128]
    const float* __restrict__ cnt,     // [N_EDGES]
    const float* __restrict__ he,      // hyperedge [N_EDGES,128]
    const v2f*   __restrict__ bswz,    // pre-swizzled W_upd fragments (64 k-steps)
    const float* __restrict__ bias,    // [128]
    float* __restrict__ out)           // [N_EDGES,128]
{
    const int lane    = threadIdx.x & 31;
    const int wave    = threadIdx.x >> 5;
    const int rowbase = (blockIdx.x * 8 + wave) * 16;
    const int m       = lane & 15;
    const int hi      = lane >> 4;

    v8f acc[8];
#pragma unroll
    for (int t = 0; t < 8; ++t) acc[t] = 0.0f;

    const int rowA = min(rowbase + m, N_EDGES - 1);
    const float rcp = 1.0f / fmaxf(cnt[rowA], 1.0f);    // scatter-mean divisor
    const float* __restrict__ arow0 = agg_sum + (long)rowA * D_FEAT;
    const float* __restrict__ arow1 = he      + (long)rowA * D_FEAT;

    // k in [0,128): mean-normalized aggregation (read-once -> NT)
    for (int kk = 0; kk < 32; ++kk) {
        const int k0 = 4 * kk + 2 * hi;
        v2f a = __builtin_nontemporal_load((const v2f*)(arow0 + k0));
        a.x *= rcp; a.y *= rcp;
        const v2f* __restrict__ bp = bswz + (size_t)(kk * 8) * 32 + lane;
#pragma unroll
        for (int nt = 0; nt < 8; ++nt) {
            v2f b = bp[nt * 32];
            acc[nt] = __builtin_amdgcn_wmma_f32_16x16x4_f32(
                false, a, false, b, (short)0, acc[nt], false, false);
        }
    }
    // k in [128,256): hyperedge features (read-once -> NT)
    for (int kk = 32; kk < 64; ++kk) {
        const int k0 = 4 * (kk - 32) + 2 * hi;
        v2f a = __builtin_nontemporal_load((const v2f*)(arow1 + k0));
        const v2f* __restrict__ bp = bswz + (size_t)(kk * 8) * 32 + lane;
#pragma unroll
        for (int nt = 0; nt < 8; ++nt) {
            v2f b = bp[nt * 32];
            acc[nt] = __builtin_amdgcn_wmma_f32_16x16x4_f32(
                false, a, false, b, (short)0, acc[nt], false, false);
        }
    }

    float bv[8];
#pragma unroll
    for (int nt = 0; nt < 8; ++nt) bv[nt] = bias[nt * 16 + m];

    // bias + relu, accumulate per-row sum of squares
    float ss[8];
#pragma unroll
    for (int j = 0; j < 8; ++j) {
        float s = 0.0f;
#pragma unroll
        for (int nt = 0; nt < 8; ++nt) {
            float v = fmaxf(acc[nt][j] + bv[nt], 0.0f);
            acc[nt][j] = v;
            s += v * v;
        }
        ss[j] = s;
    }
    // reduce across the 16 lanes sharing each row (xor masks < 16 stay in the
    // half-wave that owns the row per the C/D layout)
#pragma unroll
    for (int j = 0; j < 8; ++j) {
        float s = ss[j];
        s += __shfl_xor(s, 1, 32);
        s += __shfl_xor(s, 2, 32);
        s += __shfl_xor(s, 4, 32);
        s += __shfl_xor(s, 8, 32);
        ss[j] = s;
    }

#pragma unroll
    for (int j = 0; j < 8; ++j) {
        const int row = rowbase + j + 8 * hi;
        if (row < N_EDGES) {
            const float inv = 1.0f / fmaxf(sqrtf(ss[j]), 1e-12f);
#pragma unroll
            for (int nt = 0; nt < 8; ++nt) {
                __builtin_nontemporal_store(acc[nt][j] * inv,
                                            &out[(long)row * D_FEAT + nt * 16 + m]);
            }
        }
    }
}

// ---------------------------------------------------------------------------
extern "C" void kernel_launch(void* const* d_in, const int* in_sizes, int n_in,
                              void* d_out, int out_size, void* d_ws, size_t ws_size,
                              hipStream_t stream) {
    const float*     hyperedge  = (const float*)d_in[0];      // [E,128]
    const float*     hyper_node = (const float*)d_in[1];      // [N,128]
    const long long* aff        = (const long long*)d_in[2];  // [2,N] int64
    const float*     W_v2e      = (const float*)d_in[3];      // [128,128]
    const float*     b_v2e      = (const float*)d_in[4];      // [128]
    const float*     W_upd      = (const float*)d_in[5];      // [256,128]
    const float*     b_upd      = (const float*)d_in[6];      // [128]
    float*           out        = (float*)d_out;

    float* agg_sum = (float*)d_ws;                            // E*128 floats
    float* cnt     = agg_sum + (size_t)N_EDGES * D_FEAT;      // E floats
    float* bswz1   = cnt + N_EDGES;                           // NSWZ1 float2
    float* bswz2   = bswz1 + (size_t)NSWZ1 * 2;               // NSWZ2 float2

    zero_ws_kernel<<<1024, 256, 0, stream>>>(agg_sum, NZERO);
    swizzle_w_kernel<<<(NSWZ1 + NSWZ2 + 255) / 256, 256, 0, stream>>>(
        W_v2e, W_upd, bswz1, bswz2);

    gemm1_scatter_kernel<<<(N_NODES + 127) / 128, 256, 0, stream>>>(
        hyper_node, (const v2f*)bswz1, b_v2e, aff, agg_sum, cnt);

    gemm2_norm_kernel<<<(N_EDGES + 127) / 128, 256, 0, stream>>>(
        agg_sum, cnt, hyperedge, (const v2f*)bswz2, b_upd, out);
}